// EquivariantDense_19353122636424
// MI455X (gfx1250) — compile-verified
//
#include <hip/hip_runtime.h>

// EquivariantDense: out[b,o] = sum_j w[b,j,o] * y[b,j],
// y[b,j] = 4-term rotated gather of x[b].  B=64, J=4096, OUT=15.

typedef __attribute__((ext_vector_type(2))) float v2f;
typedef __attribute__((ext_vector_type(8))) float v8f;

#define NB    64
#define JTOT  4096      // 8*8*64
#define KROT  4
#define NOUT  15
#define WSTR  16        // padded partial stride in workspace

template<int SPLIT>
__global__ __launch_bounds__(256) void equiv_main(
    const float* __restrict__ x,   // (64, 16384)
    const float* __restrict__ w,   // (64, 4096, 15)
    float* __restrict__ dst)       // SPLIT>1: partials (stride WSTR); else out
{
    constexpr int JPER = JTOT / SPLIT;             // j's per block
    constexpr int DSTR = (SPLIT == 1) ? NOUT : WSTR;

    __shared__ float ysh[JPER];
    __shared__ float part[8][16];

    const int blk = blockIdx.x;
    const int b   = blk / SPLIT;
    const int s   = blk - b * SPLIT;
    const int j0base = s * JPER;
    const int tid = threadIdx.x;

    const float* __restrict__ xb = x + b * (JTOT * KROT);  // 16384 floats
    const float* __restrict__ wb = w + b * (JTOT * NOUT);  // 61440 floats

    // ---- Phase 1: y[j] = sum of 4 inverse-rotated input gathers ----------
    #pragma unroll
    for (int jj = tid; jj < JPER; jj += 256) {
        const int j  = j0base + jj;
        const int h  = (j >> 9) & 7;
        const int wd = (j >> 6) & 7;
        const int c  =  j       & 63;
        float v =
            xb[((( h      * 8 +  wd     ) * 64 + c) << 2) + 0] +
            xb[((((7 - wd)* 8 +  h      ) * 64 + c) << 2) + 1] +
            xb[((((7 - h) * 8 + (7 - wd)) * 64 + c) << 2) + 2] +
            xb[((( wd     * 8 + (7 - h) ) * 64 + c) << 2) + 3];
        ysh[jj] = v;
    }
    __syncthreads();

    // ---- Phase 2: WMMA f32 16x16x4 matvec over this wave's j-range -------
    // A[m,k] = w[j0+k, m] * y[j0+k]  (documented A layout: M=lane&15,
    // K = (lane>>4)*2 + vgpr).  B = all-ones (layout independent), so
    // D[m,n] accumulates sum_k w[j0+k,m]*y[j0+k] in every column n.
    // Two independent accumulator chains break the WMMA D->C serial dep.
    const int wave = tid >> 5;
    const int lane = tid & 31;
    const int m    = lane & 15;
    const int mo   = (m < NOUT) ? m : (NOUT - 1);  // clamp pad row (no OOB)
    const int kb   = (lane >> 4) * 2;              // 0 or 2

    v8f acc0 = {};
    v8f acc1 = {};
    const v2f ones = {1.0f, 1.0f};

    constexpr int JWAVE = JPER >> 3;   // j's handled by this wave
    const int jlo = wave * JWAVE;      // local index into ysh

    #pragma unroll 2
    for (int q = 0; q < JWAVE; q += 8) {
        // chunk A (even)
        {
            const int jl = jlo + q;
            const int jg = j0base + jl;
            const float y0 = ysh[jl + kb];
            const float y1 = ysh[jl + kb + 1];
            const float a0 = wb[(jg + kb)     * NOUT + mo] * y0;
            const float a1 = wb[(jg + kb + 1) * NOUT + mo] * y1;
            v2f a = { a0, a1 };
            acc0 = __builtin_amdgcn_wmma_f32_16x16x4_f32(
                false, a, false, ones, (short)0, acc0, false, false);
        }
        // chunk B (odd)
        {
            const int jl = jlo + q + 4;
            const int jg = j0base + jl;
            const float y0 = ysh[jl + kb];
            const float y1 = ysh[jl + kb + 1];
            const float a0 = wb[(jg + kb)     * NOUT + mo] * y0;
            const float a1 = wb[(jg + kb + 1) * NOUT + mo] * y1;
            v2f a = { a0, a1 };
            acc1 = __builtin_amdgcn_wmma_f32_16x16x4_f32(
                false, a, false, ones, (short)0, acc1, false, false);
        }
    }

    v8f acc = acc0 + acc1;

    // ---- Phase 3: extract column 0 (all columns equal) & cross-wave add --
    // C/D layout: lanes 0-15 hold M=vgpr (0..7), lanes 16-31 hold M=8+vgpr.
    if (lane == 0) {
        #pragma unroll
        for (int r = 0; r < 8; ++r) part[wave][r] = acc[r];
    } else if (lane == 16) {
        #pragma unroll
        for (int r = 0; r < 8; ++r) part[wave][8 + r] = acc[r];
    }
    __syncthreads();

    if (tid < NOUT) {
        float sum = 0.0f;
        #pragma unroll
        for (int wv = 0; wv < 8; ++wv) sum += part[wv][tid];
        dst[blk * DSTR + tid] = sum;
    }
}

__global__ __launch_bounds__(256) void equiv_reduce(
    const float* __restrict__ ws, float* __restrict__ out, int split)
{
    const int t = blockIdx.x * blockDim.x + threadIdx.x;
    if (t >= NB * NOUT) return;
    const int b = t / NOUT;
    const int m = t - b * NOUT;
    float sum = 0.0f;
    for (int q = 0; q < split; ++q)
        sum += ws[(b * split + q) * WSTR + m];
    out[t] = sum;
}

extern "C" void kernel_launch(void* const* d_in, const int* in_sizes, int n_in,
                              void* d_out, int out_size, void* d_ws, size_t ws_size,
                              hipStream_t stream) {
    (void)in_sizes; (void)n_in; (void)out_size;
    const float* x = (const float*)d_in[0];   // (64, 16384) f32
    const float* w = (const float*)d_in[1];   // (64, 8, 8, 64, 15) f32
    float* out = (float*)d_out;               // 960 f32

    constexpr int SPLIT = 4;
    const size_t need = (size_t)NB * SPLIT * WSTR * sizeof(float);
    if (ws_size >= need) {
        float* ws = (float*)d_ws;
        equiv_main<SPLIT><<<NB * SPLIT, 256, 0, stream>>>(x, w, ws);
        equiv_reduce<<<(NB * NOUT + 255) / 256, 256, 0, stream>>>(ws, out, SPLIT);
    } else {
        // Workspace too small: single block per batch, write final directly.
        equiv_main<1><<<NB, 256, 0, stream>>>(x, w, out);
    }
}